// PointPillarScatter_17051020165835
// MI455X (gfx1250) — compile-verified
//
#include <hip/hip_runtime.h>

// PointPillarScatter for MI455X (gfx1250).
// Pure data movement: 219.4 MB of output stores dominates -> optimize as a
// single-pass, fully-coalesced gather with non-temporal (streaming) stores.

typedef float v4f __attribute__((ext_vector_type(4)));

#define NXD   432
#define NYD   496
#define CD    64
#define BD    4
#define PLANE (NXD * NYD)        // 214272 elements per (b,c) plane
#define MAPN  (BD * PLANE)       // 857088 map entries (3.43 MB)
#define XQ    (NXD / 4)          // 108 float4-groups per row

// ---------------------------------------------------------------------------
// Kernel 1: map[i] = -1 for all i. Exact grid (MAPN/4 int4 stores), no checks.
// ---------------------------------------------------------------------------
__global__ void pps_init_map(int* __restrict__ map) {
    int i = blockIdx.x * blockDim.x + threadIdx.x;   // one int4 per thread
    int4 m = make_int4(-1, -1, -1, -1);
    reinterpret_cast<int4*>(map)[i] = m;
}

// ---------------------------------------------------------------------------
// Kernel 2: map[(b*NY + y)*NX + x] = p for each pillar p.
// Coords are int32 triples (x, y, b) — JAX downcasts int64 -> int32.
// Flat indices are unique per the reference's sampling-without-replacement.
// ---------------------------------------------------------------------------
__global__ void pps_fill_map(const int* __restrict__ coords,
                             int* __restrict__ map, int P) {
    int p = blockIdx.x * blockDim.x + threadIdx.x;
    if (p >= P) return;
    int x = coords[3 * p + 0];
    int y = coords[3 * p + 1];
    int b = coords[3 * p + 2];
    map[(b * NYD + y) * NXD + x] = p;
}

// ---------------------------------------------------------------------------
// Kernel 3: gather. Thread s owns 4 consecutive x of one (b, y) and writes all
// 64 channels: out[((b*C + c)*NY + y)*NX + x..x+3] per iteration.
// A wave32 covers 128 consecutive x -> each c-iteration stores 512 contiguous
// bytes (32 lanes x float4). Stores are non-temporal: the 219 MB output stream
// must not evict the 13.6 MB of hot map+feature data from the 192 MB L2.
// Occupied lanes prefetch their 256 B feature row (global_prefetch_b8), then
// gather with clamped pointers so the inner loop is load + v_cndmask, branch-
// free and never out of bounds.
// ---------------------------------------------------------------------------
__global__ void pps_scatter(const float* __restrict__ feat,
                            const int* __restrict__ map,
                            float* __restrict__ out) {
    int s  = blockIdx.x * blockDim.x + threadIdx.x;  // [0, B*NY*XQ)
    int xq = s % XQ;
    int t  = s / XQ;
    int y  = t % NYD;
    int b  = t / NYD;

    int spatial = (b * NYD + y) * NXD + (xq << 2);   // 16B-aligned map offset
    int4 p4 = reinterpret_cast<const int4*>(map)[spatial >> 2];

    float* outp = out + (size_t)b * (CD * (size_t)PLANE)
                      + (size_t)y * NXD + (size_t)(xq << 2);

    if ((p4.x & p4.y & p4.z & p4.w) < 0) {
        // All four positions empty (95% case): pure streaming zero stores.
        v4f z = (v4f)0.0f;
#pragma unroll 8
        for (int c = 0; c < CD; ++c) {
            __builtin_nontemporal_store(z, reinterpret_cast<v4f*>(outp));
            outp += PLANE;
        }
    } else {
        bool v0 = p4.x >= 0, v1 = p4.y >= 0, v2 = p4.z >= 0, v3 = p4.w >= 0;
        // Clamp invalid rows to row 0 -> loads always in-bounds, value
        // selected out with cndmask below.
        const float* f0 = feat + (size_t)(v0 ? p4.x : 0) * CD;
        const float* f1 = feat + (size_t)(v1 ? p4.y : 0) * CD;
        const float* f2 = feat + (size_t)(v2 ? p4.z : 0) * CD;
        const float* f3 = feat + (size_t)(v3 ? p4.w : 0) * CD;
        // gfx1250 prefetch: pull each 256 B feature row toward the WGP before
        // the channel loop starts consuming it.
        __builtin_prefetch(f0, 0, 1);
        __builtin_prefetch(f1, 0, 1);
        __builtin_prefetch(f2, 0, 1);
        __builtin_prefetch(f3, 0, 1);
#pragma unroll 8
        for (int c = 0; c < CD; ++c) {
            v4f v;
            v.x = v0 ? f0[c] : 0.0f;
            v.y = v1 ? f1[c] : 0.0f;
            v.z = v2 ? f2[c] : 0.0f;
            v.w = v3 ? f3[c] : 0.0f;
            __builtin_nontemporal_store(v, reinterpret_cast<v4f*>(outp));
            outp += PLANE;
        }
    }
}

// ---------------------------------------------------------------------------
// Launch: init map (-1) -> fill map -> single-pass gather. All on `stream`,
// graph-capture safe (kernels only, no memset/malloc/sync APIs).
// ---------------------------------------------------------------------------
extern "C" void kernel_launch(void* const* d_in, const int* in_sizes, int n_in,
                              void* d_out, int out_size, void* d_ws, size_t ws_size,
                              hipStream_t stream) {
    const float* feat   = (const float*)d_in[0];   // (P, 64) fp32
    const int*   coords = (const int*)d_in[1];     // (P, 3)  int32
    int P = in_sizes[0] / CD;                      // 40000

    int* map = (int*)d_ws;                         // 3.43 MB scratch

    // MAPN/4 int4 inits: 214272 threads = 837 blocks of 256, exact.
    pps_init_map<<<(MAPN / 4) / 256, 256, 0, stream>>>(map);

    pps_fill_map<<<(P + 255) / 256, 256, 0, stream>>>(coords, map, P);

    // B*NY*XQ = 214272 threads = 837 blocks of 256, exact.
    pps_scatter<<<(BD * NYD * XQ) / 256, 256, 0, stream>>>(
        feat, map, (float*)d_out);
}